// CBAM3D_88407606821220
// MI455X (gfx1250) — compile-verified
//
#include <hip/hip_runtime.h>
#include <hip/hip_bf16.h>
#include <math.h>

typedef __attribute__((ext_vector_type(2))) float v2f;
typedef __attribute__((ext_vector_type(8))) float v8f;

#define B_   4
#define C_   64
#define S_   (64*64*64)      // 262144 spatial voxels per batch
#define HID  4
#define KS   7
#define NEG_INF_KEY 0x007FFFFFu   // order-encoded -inf

// monotone float->uint mapping so unsigned atomicMax == float max
__device__ __forceinline__ unsigned encodeOrd(float f) {
    unsigned u = __float_as_uint(f);
    return (u & 0x80000000u) ? ~u : (u | 0x80000000u);
}
__device__ __forceinline__ float decodeOrd(unsigned k) {
    unsigned u = (k & 0x80000000u) ? (k & 0x7FFFFFFFu) : ~k;
    return __uint_as_float(u);
}
__device__ __forceinline__ float sigmoidf_(float v) {
    return 1.0f / (1.0f + __expf(-v));
}

// ---------------- kernel 0: reset reduction scratch every call ----------------
__global__ void k_init(float* __restrict__ sumPool, unsigned* __restrict__ maxKey) {
    int t = threadIdx.x;
    if (t < B_ * C_) { sumPool[t] = 0.0f; maxKey[t] = NEG_INF_KEY; }
}

// ---------------- kernel 1: channel pooling (mean-sum + max over D*H*W) ------
// grid = B_*512 blocks, 256 threads; block handles 512 voxels of one batch.
__global__ void k_pool(const float* __restrict__ x,
                       float* __restrict__ sumPool, unsigned* __restrict__ maxKey) {
    __shared__ float sS[256];
    __shared__ float sM[256];
    int tid = threadIdx.x;
    int c = tid & 63, g = tid >> 6;                 // 4 groups x 64 channels
    int b = blockIdx.x >> 9;                        // /512
    int chunk = blockIdx.x & 511;
    size_t base = ((size_t)b * S_ + (size_t)chunk * 512) * C_;
    float s = 0.0f, m = -INFINITY;
    #pragma unroll 4
    for (int i = 0; i < 128; ++i) {
        float v = x[base + (size_t)(g + i * 4) * C_ + c];
        s += v;
        m = fmaxf(m, v);
    }
    sS[tid] = s; sM[tid] = m;
    __syncthreads();
    if (tid < 64) {
        s = sS[tid] + sS[tid + 64] + sS[tid + 128] + sS[tid + 192];
        m = fmaxf(fmaxf(sM[tid], sM[tid + 64]), fmaxf(sM[tid + 128], sM[tid + 192]));
        atomicAdd(&sumPool[b * C_ + tid], s);
        atomicMax(&maxKey[b * C_ + tid], encodeOrd(m));
    }
}

// ---------------- kernel 2: channel-attention MLP via V_WMMA_F32_16X16X4_F32 --
// One wave32. A[16x64]: rows 0-3 = avg-pool(b), rows 4-7 = max-pool(b), rest 0.
// All operands staged zero-padded in LDS so the WMMA loop is branch-free
// (EXEC must be all 1s around WMMA anyway).
__global__ void k_mlp(const float* __restrict__ sumPool, const unsigned* __restrict__ maxKey,
                      const float* __restrict__ w1, const float* __restrict__ b1,
                      const float* __restrict__ w2, const float* __restrict__ b2,
                      float* __restrict__ ca) {
    __shared__ float sPool[16][C_];   // A matrix, rows 8-15 zero
    __shared__ float sW1[C_][16];     // B matrix layer 1, cols 4-15 zero
    __shared__ float hLds[16][HID];   // hidden activations, row-major
    const float invS = 1.0f / (float)S_;
    int lane = threadIdx.x;
    int m  = lane & 15;        // M (A rows / C rows)
    int kh = lane >> 4;        // K-half select per ISA 16x4 f32 layout
    int n  = lane & 15;        // N (B/C cols)

    // stage A (avg rows 0-3, max rows 4-7, zero pad) and padded w1 into LDS
    for (int i = lane; i < 16 * C_; i += 32) {
        int row = i >> 6, col = i & 63;
        float v = 0.0f;
        if (row < B_)            v = sumPool[row * C_ + col] * invS;
        else if (row < 2 * B_)   v = decodeOrd(maxKey[(row - B_) * C_ + col]);
        sPool[row][col] = v;
    }
    for (int i = lane; i < C_ * 16; i += 32) {
        int k = i >> 4, col = i & 15;
        sW1[k][col] = (col < HID) ? w1[k * HID + col] : 0.0f;
    }
    __syncthreads();

    // C init = bias b1 broadcast along M (cols >= HID zero-padded)
    v8f acc;
    float c0 = (n < HID) ? b1[n] : 0.0f;
    #pragma unroll
    for (int r = 0; r < 8; ++r) acc[r] = c0;

    #pragma unroll
    for (int k0 = 0; k0 < C_; k0 += 4) {
        v2f a, bb;
        #pragma unroll
        for (int j = 0; j < 2; ++j) {
            int k = k0 + 2 * kh + j;          // A/B: VGPR j holds K = j + 2*kh
            a[j]  = sPool[m][k];
            bb[j] = sW1[k][n];
        }
        acc = __builtin_amdgcn_wmma_f32_16x16x4_f32(false, a, false, bb,
                                                    (short)0, acc, false, false);
    }

    // ReLU; stash h row-major to LDS (C/D layout: VGPR r -> M=r / r+8)
    #pragma unroll
    for (int r = 0; r < 8; ++r) {
        int M = kh ? (r + 8) : r;
        float hv = fmaxf(acc[r], 0.0f);
        if (n < HID) hLds[M][n] = hv;
    }
    __syncthreads();

    // A2 = h in 16x4 A layout
    v2f a2;
    #pragma unroll
    for (int j = 0; j < 2; ++j) a2[j] = hLds[m][2 * kh + j];

    #pragma unroll
    for (int t = 0; t < 4; ++t) {
        int n0 = t * 16;
        v2f bv;
        #pragma unroll
        for (int j = 0; j < 2; ++j) bv[j] = w2[(2 * kh + j) * C_ + n0 + n];
        v8f c2;
        float cb = b2[n0 + n];
        #pragma unroll
        for (int r = 0; r < 8; ++r) c2[r] = cb;
        v8f d = __builtin_amdgcn_wmma_f32_16x16x4_f32(false, a2, false, bv,
                                                      (short)0, c2, false, false);
        if (lane < 16) {   // rows 0..7 live in lanes 0-15, VGPRs 0..7
            #pragma unroll
            for (int r = 0; r < B_; ++r)
                ca[r * C_ + n0 + lane] = sigmoidf_(d[r]) + sigmoidf_(d[r + B_]);
        }
    }
}

// ---------------- kernel 3: spatial mean/max over C of (x * ca) --------------
// No materialization of channel_refined: recompute it in kernel 4 instead.
// 1024 blocks per batch, 256 threads; one voxel (64 ch) per thread, float4 I/O.
__global__ void k_spool(const float* __restrict__ x, const float* __restrict__ ca,
                        float* __restrict__ avg_sp, float* __restrict__ max_sp) {
    __shared__ __align__(16) float sca[C_];
    int b = blockIdx.x >> 10;
    if (threadIdx.x < C_) sca[threadIdx.x] = ca[b * C_ + threadIdx.x];
    __syncthreads();
    size_t idx = (size_t)blockIdx.x * blockDim.x + threadIdx.x;   // global voxel id
    const float4* x4 = (const float4*)(x + idx * C_);
    const float4* c4 = (const float4*)sca;
    float s = 0.0f, mx = -INFINITY;
    #pragma unroll
    for (int i = 0; i < C_ / 4; ++i) {
        float4 v = x4[i];
        float4 g = c4[i];
        v.x *= g.x; v.y *= g.y; v.z *= g.z; v.w *= g.w;
        s += v.x + v.y + v.z + v.w;
        mx = fmaxf(mx, fmaxf(fmaxf(v.x, v.y), fmaxf(v.z, v.w)));
    }
    avg_sp[idx] = s * (1.0f / (float)C_);
    max_sp[idx] = mx;
}

// ---------------- kernel 4: 7^3x2 conv (SAME) -> sigmoid; out = x*ca*sa ------
// pooled maps (8 MB) stay resident in the 192 MB L2 across their 343x reuse.
__global__ void k_conv(const float* __restrict__ x, const float* __restrict__ ca,
                       const float* __restrict__ avg_sp, const float* __restrict__ max_sp,
                       const float* __restrict__ cw, float* __restrict__ out) {
    __shared__ float w[KS * KS * KS * 2];
    __shared__ __align__(16) float sca[C_];
    int b = blockIdx.x >> 10;
    for (int i = threadIdx.x; i < KS * KS * KS * 2; i += blockDim.x) w[i] = cw[i];
    if (threadIdx.x < C_) sca[threadIdx.x] = ca[b * C_ + threadIdx.x];
    __syncthreads();
    size_t idx = (size_t)blockIdx.x * blockDim.x + threadIdx.x;
    int p = (int)(idx & (size_t)(S_ - 1));
    int d = p >> 12, h = (p >> 6) & 63, wv = p & 63;
    const float* A = avg_sp + (size_t)b * S_;
    const float* M = max_sp + (size_t)b * S_;
    float acc = 0.0f;
    for (int kd = 0; kd < KS; ++kd) {
        int zd = d + kd - 3; if ((unsigned)zd >= 64u) continue;
        for (int khh = 0; khh < KS; ++khh) {
            int zh = h + khh - 3; if ((unsigned)zh >= 64u) continue;
            int rowBase = (zd * 64 + zh) * 64;
            int wBase = ((kd * KS + khh) * KS) * 2;
            #pragma unroll
            for (int kw = 0; kw < KS; ++kw) {
                int zw = wv + kw - 3; if ((unsigned)zw >= 64u) continue;
                int q = rowBase + zw;
                acc = fmaf(A[q], w[wBase + kw * 2], acc);
                acc = fmaf(M[q], w[wBase + kw * 2 + 1], acc);
            }
        }
    }
    float sa = sigmoidf_(acc);
    const float4* x4 = (const float4*)(x + idx * C_);
    const float4* c4 = (const float4*)sca;
    float4* o4 = (float4*)(out + idx * C_);
    #pragma unroll
    for (int i = 0; i < C_ / 4; ++i) {
        float4 v = x4[i];
        float4 g = c4[i];
        v.x *= g.x * sa; v.y *= g.y * sa; v.z *= g.z * sa; v.w *= g.w * sa;
        o4[i] = v;
    }
}

extern "C" void kernel_launch(void* const* d_in, const int* in_sizes, int n_in,
                              void* d_out, int out_size, void* d_ws, size_t ws_size,
                              hipStream_t stream) {
    const float* x  = (const float*)d_in[0];
    const float* w1 = (const float*)d_in[1];
    const float* b1 = (const float*)d_in[2];
    const float* w2 = (const float*)d_in[3];
    const float* b2 = (const float*)d_in[4];
    const float* cw = (const float*)d_in[5];
    float* out = (float*)d_out;

    // workspace layout (floats): [0,256) sumPool | [256,512) maxKey(u32)
    // | [512,768) ca | [1024, 1024+B*S) avg_sp | [... + B*S) max_sp
    float*    ws      = (float*)d_ws;
    float*    sumPool = ws;
    unsigned* maxKey  = (unsigned*)(ws + 256);
    float*    ca      = ws + 512;
    float*    avg_sp  = ws + 1024;
    float*    max_sp  = ws + 1024 + (size_t)B_ * S_;

    k_init <<<1, 256, 0, stream>>>(sumPool, maxKey);
    k_pool <<<B_ * 512, 256, 0, stream>>>(x, sumPool, maxKey);
    k_mlp  <<<1, 32, 0, stream>>>(sumPool, maxKey, w1, b1, w2, b2, ca);
    k_spool<<<B_ * 1024, 256, 0, stream>>>(x, ca, avg_sp, max_sp);
    k_conv <<<B_ * 1024, 256, 0, stream>>>(x, ca, avg_sp, max_sp, cw, out);
}